// CTCLossMPS_3856880632300
// MI455X (gfx1250) — compile-verified
//
#include <hip/hip_runtime.h>
#include <cstdint>

#define NEGF (-1e30f)

#ifndef __has_builtin
#define __has_builtin(x) 0
#endif

typedef int v4i_t __attribute__((vector_size(4 * sizeof(int))));

// ---------------------------------------------------------------------------
// CDNA5 async global->LDS copy (ASYNCcnt-tracked), 16 bytes per lane.
// Builtin signature (probe-confirmed via hipcc diagnostic): arg0 = v4i global
// ptr, arg1 = v4i LDS ptr, then two integer immediates (offset, cpol).
// Generic->AS3 cast via 32-bit truncation is valid on AMDGCN: generic LDS
// pointers carry the LDS byte offset in their low 32 bits.
// ---------------------------------------------------------------------------
__device__ __forceinline__ void async_load_b128(const void* gsrc, void* ldst) {
#if __has_builtin(__builtin_amdgcn_global_load_async_to_lds_b128)
  __builtin_amdgcn_global_load_async_to_lds_b128(
      (__attribute__((address_space(1))) v4i_t*)(uintptr_t)gsrc,
      (__attribute__((address_space(3))) v4i_t*)(unsigned)(uintptr_t)ldst,
      0, 0);
#else
  unsigned lds_addr = (unsigned)(uintptr_t)ldst;
  unsigned long long gaddr = (unsigned long long)(uintptr_t)gsrc;
  asm volatile("global_load_async_to_lds_b128 %0, %1, off"
               :: "v"(lds_addr), "v"(gaddr)
               : "memory");
#endif
}

template <int N>
__device__ __forceinline__ void wait_asynccnt_le() {
#if __has_builtin(__builtin_amdgcn_s_wait_asynccnt)
  __builtin_amdgcn_s_wait_asynccnt(N);
#else
  asm volatile("s_wait_asynccnt %0" :: "n"(N) : "memory");
#endif
}

// ---------------------------------------------------------------------------
// One workgroup per batch element b. Thread l owns CTC lattice cell l.
// Row log_probs[t,b,:] (C floats = 4KB) is streamed into a DEPTH-deep LDS
// ring buffer by async-to-LDS loads (8 rows in flight => ~1us of HBM-latency
// cover per wave); the alpha recurrence ping-pongs between two LDS arrays
// with 2 guard slots at the front (indices l-1, l-2 need no branches).
// ---------------------------------------------------------------------------
__global__ void __launch_bounds__(512)
ctc_alpha_kernel(const float* __restrict__ logp,      // [T,B,C] log-softmaxed
                 const int*   __restrict__ targets,   // [B,S]
                 const int*   __restrict__ in_len,    // [B]
                 const int*   __restrict__ tgt_len,   // [B]
                 float*       __restrict__ losses,    // [B] out (NLL)
                 int T, int B, int C, int S)
{
  constexpr int DEPTH = 8;          // async pipeline depth (ring slots)
  constexpr int MAXC  = 1024;       // classes (reference setup)
  constexpr int MAXL2 = 2 * 128 + 1 + 2;  // L + 2 guard slots

  __shared__ __align__(16) float row[DEPTH][MAXC];   // 32 KB of 320 KB LDS
  __shared__ float bufA[MAXL2];
  __shared__ float bufB[MAXL2];

  const int b      = blockIdx.x;
  const int tid    = threadIdx.x;
  const int L      = 2 * S + 1;
  const int nChunk = C >> 2;        // 16B chunks per row (256)

  // Per-cell static parameters (thread tid owns lattice index l = tid).
  const int  l    = tid;
  const bool inL  = (l < L);
  const int  Tin  = in_len[b];
  const int  Lb   = 2 * tgt_len[b] + 1;
  const bool validl = inL && (l < Lb);
  int  extl = 0;                    // expanded label: blank(0) at even l
  bool skip = false;                // s-2 -> s transition allowed
  if (inL && (l & 1)) {
    const int half = l >> 1;
    extl = targets[b * S + half];
    skip = (l >= 3) && (extl != targets[b * S + half - 1]);
  }

  if (tid < 2) { bufA[tid] = NEGF; bufB[tid] = NEGF; }  // guards

  float* cur = bufA + 2;
  float* nxt = bufB + 2;

  const float*  rowbase = logp + (size_t)b * C;
  const size_t  tstride = (size_t)B * C;

  // Prologue: fill the async pipeline.
  for (int t = 0; t < DEPTH && t < T; ++t) {
    if (tid < nChunk)
      async_load_b128(rowbase + (size_t)t * tstride + tid * 4,
                      &row[t & (DEPTH - 1)][tid * 4]);
  }

  for (int t = 0; t < T; ++t) {
    // Async loads complete in-order: cnt <= DEPTH-1 retires row t exactly.
    wait_asynccnt_le<DEPTH - 1>();
    __syncthreads();   // all waves' row portions + prev alpha writes visible

    const float* r = row[t & (DEPTH - 1)];
    if (inL) {
      const float lp = r[extl];     // LDS gather of emission log-prob
      float out;
      if (t == 0) {
        out = (l <= 1) ? lp : NEGF;
      } else if (t < Tin) {
        const float a  = cur[l];
        const float a1 = cur[l - 1];
        const float a2 = skip ? cur[l - 2] : NEGF;
        const float m   = fmaxf(a, fmaxf(a1, a2));
        const float lse = m + __logf(__expf(a  - m) +
                                     __expf(a1 - m) +
                                     __expf(a2 - m));
        out = lse + lp;
      } else {
        out = cur[l];               // frozen past input length
      }
      nxt[l] = validl ? out : NEGF;
    }
    __syncthreads();   // slot (t & 7) fully consumed; alpha writes complete

    const int tn = t + DEPTH;       // refill the freed slot
    if (tn < T && tid < nChunk)
      async_load_b128(rowbase + (size_t)tn * tstride + tid * 4,
                      &row[t & (DEPTH - 1)][tid * 4]);

    float* tmp = cur; cur = nxt; nxt = tmp;
  }

  if (tid == 0) {
    const float last = cur[Lb - 1];
    const float prev = cur[Lb - 2];
    const float m    = fmaxf(last, prev);
    const float fin  = m + __logf(__expf(last - m) + __expf(prev - m));
    losses[b] = -fin;
  }
}

// Deterministic mean reduction (no float atomics -> bitwise reproducible).
__global__ void ctc_finalize_kernel(const float* __restrict__ losses,
                                    const int*   __restrict__ tgt_len,
                                    float* __restrict__ out, int B)
{
  if (threadIdx.x == 0 && blockIdx.x == 0) {
    float s = 0.0f;
    for (int bb = 0; bb < B; ++bb)
      s += losses[bb] / (float)tgt_len[bb];
    out[0] = s / (float)B;
  }
}

extern "C" void kernel_launch(void* const* d_in, const int* in_sizes, int n_in,
                              void* d_out, int out_size, void* d_ws, size_t ws_size,
                              hipStream_t stream) {
  const float* logp    = (const float*)d_in[0];   // [T,B,C] f32
  const int*   targets = (const int*)  d_in[1];   // [B,S] i32
  const int*   in_len  = (const int*)  d_in[2];   // [B] i32
  const int*   tgt_len = (const int*)  d_in[3];   // [B] i32

  const int B = in_sizes[2];
  const int S = in_sizes[1] / B;
  const int C = 1024;  // num classes per reference setup
  const int T = (int)((long long)in_sizes[0] / ((long long)B * C));
  const int L = 2 * S + 1;

  int threads = (L > (C >> 2)) ? L : (C >> 2);
  threads = ((threads + 31) / 32) * 32;           // 288 = 9 waves (wave32)

  float* losses = (float*)d_ws;                   // B floats of scratch

  ctc_alpha_kernel<<<B, threads, 0, stream>>>(
      logp, targets, in_len, tgt_len, losses, T, B, C, S);
  ctc_finalize_kernel<<<1, 32, 0, stream>>>(losses, tgt_len, (float*)d_out, B);
}